// ESN_4544075399504
// MI455X (gfx1250) — compile-verified
//
#include <hip/hip_runtime.h>
#include <math.h>

// ---------------- CDNA5 / gfx1250 ESN persistent kernel ----------------
// 16 blocks (1 per WGP) x 256 threads (8 wave32). W partitioned row-wise,
// pinned in LDS (staged by the Tensor Data Mover). fp32 WMMA 16x16x4 for the
// 64x1024 per-WGP matvec. Per-step state broadcast via TDM global->LDS.

typedef __attribute__((ext_vector_type(2))) float v2f;
typedef __attribute__((ext_vector_type(8))) float v8f;

#if defined(__has_builtin)
#  if __has_builtin(__builtin_amdgcn_wmma_f32_16x16x4_f32)
#    define HAVE_WMMA_F32 1
#  endif
#  if __has_builtin(__builtin_amdgcn_tensor_load_to_lds) && \
      __has_builtin(__builtin_amdgcn_s_wait_tensorcnt)
#    define HAVE_TDM 1
#  endif
#endif
#ifndef HAVE_WMMA_F32
#  define HAVE_WMMA_F32 0
#endif
#ifndef HAVE_TDM
#  define HAVE_TDM 0
#endif

constexpr int T_STEPS = 32768;
constexpr int NRES = 1024;
constexpr int NIN  = 64;
constexpr int NOUT = 64;
constexpr int NCAT = NRES + NIN;     // 1088
constexpr int NB   = 16;             // blocks == WGPs used
constexpr int NT   = 256;            // threads per block (8 wave32)
constexpr int ROWS = NRES / NB;      // 64 reservoir rows per block
constexpr int YROWS = NOUT / NB;     // 4 readout rows per block
constexpr int WSTRIDE = 1028;        // padded W row stride (1 DW pad / 256 DW)
constexpr int SSTRIDE = 65;          // padded stride for skinny matrices

constexpr int SMEM_FLOATS =
    ROWS * WSTRIDE            // Ws       263168 B
  + ROWS * SSTRIDE            // W_in_s    16640 B
  + ROWS * SSTRIDE            // W_fb_s    16640 B
  + YROWS * NCAT              // W_out_s   17408 B
  + NRES                      // state      4096 B
  + NIN + NOUT + ROWS         // u, yprev, bias
  + 2 * ROWS                  // partials
  + NT;                       // reduction
// total = 320256 bytes < 320 KB WGP LDS

// --- device-wide sense barrier for the persistent kernel (16 resident blocks)
__device__ __forceinline__ void grid_barrier(unsigned* cnt, unsigned* gen) {
  __syncthreads();
  if (threadIdx.x == 0) {
    __threadfence();
    unsigned g = __hip_atomic_load(gen, __ATOMIC_RELAXED, __HIP_MEMORY_SCOPE_AGENT);
    unsigned arrived =
        __hip_atomic_fetch_add(cnt, 1u, __ATOMIC_ACQ_REL, __HIP_MEMORY_SCOPE_AGENT);
    if (arrived == (unsigned)(NB - 1)) {
      __hip_atomic_store(cnt, 0u, __ATOMIC_RELAXED, __HIP_MEMORY_SCOPE_AGENT);
      __hip_atomic_fetch_add(gen, 1u, __ATOMIC_RELEASE, __HIP_MEMORY_SCOPE_AGENT);
    } else {
      while (__hip_atomic_load(gen, __ATOMIC_ACQUIRE, __HIP_MEMORY_SCOPE_AGENT) == g) {
        __builtin_amdgcn_s_sleep(1);
      }
    }
    __threadfence();
  }
  __syncthreads();
}

#if HAVE_TDM
typedef __attribute__((ext_vector_type(4))) unsigned u32x4;
typedef __attribute__((ext_vector_type(8))) int i32x8;
typedef __attribute__((ext_vector_type(4))) int i32x4;

// Issue a 2-D TDM load: global (td0 x td1 elems, 4B, row stride s0) -> LDS.
// If padded!=0, insert 1 DWORD of LDS pad per 256 DWORDs stored
// (=> row stride 1028 floats, element k lands at k + (k>>8) within a row).
// CPOL must be an immediate (builtin requires a constant integer).
template <int CPOL>
__device__ __forceinline__ void tdm_load_2d(unsigned lds_off, const void* gptr,
                                            unsigned td0, unsigned td1,
                                            unsigned tile0, unsigned tile1,
                                            unsigned s0, int padded) {
  unsigned long long ga = (unsigned long long)gptr;
  u32x4 g0;
  g0.x = 1u;                                     // count=1, user descriptor
  g0.y = lds_off;                                // D#.lds_addr (bytes)
  g0.z = (unsigned)ga;                           // global_addr[31:0]
  g0.w = (unsigned)((ga >> 32) & 0x01FFFFFFu)    // global_addr[56:32]
       | 0x80000000u;                            // type=2 ("image")
  i32x8 g1;
  g1[0] = (int)((2u << 16) |                     // data_size = 4B
                (padded ? ((1u << 20)            // pad_enable
                         | (7u << 22)            // pad_interval = 256 DW
                         | (0u << 25)) : 0u));   // pad_amount  = 1 DW
  g1[1] = (int)((td0 & 0xFFFFu) << 16);          // tensor_dim0[15:0]
  g1[2] = (int)((td0 >> 16) | ((td1 & 0xFFFFu) << 16));  // td0 hi | td1 lo
  g1[3] = (int)((td1 >> 16) | (tile0 << 16));    // td1 hi | tile_dim0
  g1[4] = (int)(tile1 & 0xFFFFu);                // tile_dim1 (tile_dim2=0)
  g1[5] = (int)s0;                               // tensor_dim0_stride[31:0]
  g1[6] = 0;                                     // stride hi | dim1_stride lo
  g1[7] = 0;
  i32x4 z4 = {0, 0, 0, 0};                       // groups 2/3 unused (2-D)
  i32x8 z8 = {0, 0, 0, 0, 0, 0, 0, 0};
  __builtin_amdgcn_tensor_load_to_lds(g0, g1, z4, z4, z8, CPOL);
}
#endif

__global__ void esn_init_kernel(unsigned* cnt, unsigned* gen) {
  *cnt = 0u;
  *gen = 0u;
}

__global__ void __launch_bounds__(NT, 1) esn_kernel(
    const float* __restrict__ inputs,     // [T, 64]
    const float* __restrict__ W,          // [1024, 1024]
    const float* __restrict__ W_in,       // [1024, 64]
    const float* __restrict__ W_fb,       // [1024, 64]
    const float* __restrict__ W_out,      // [64, 1088]
    const float* __restrict__ noise,      // [T, 1024]
    const float* __restrict__ laststate,  // [1024]
    const float* __restrict__ lastoutput, // [64]
    float* __restrict__ out,              // [T, 64]
    float* __restrict__ stateG,           // ws: [1024]
    float* __restrict__ yG,               // ws: [64]
    unsigned* cnt, unsigned* gen)         // ws: barrier
{
  extern __shared__ float smem[];
  float* Ws    = smem;                        // [64][1028] (intra-row pads)
  float* Wins  = Ws   + ROWS * WSTRIDE;       // [64][65]
  float* Wfbs  = Wins + ROWS * SSTRIDE;       // [64][65]
  float* Wouts = Wfbs + ROWS * SSTRIDE;       // [4][1088]
  float* st    = Wouts + YROWS * NCAT;        // [1024]
  float* u_s   = st + NRES;                   // [64]
  float* yprev = u_s + NIN;                   // [64]
  float* bias  = yprev + NOUT;                // [64]
  float* part  = bias + ROWS;                 // [2][64]
  float* red   = part + 2 * ROWS;             // [256]

  const int tid = threadIdx.x;
  const int blk = blockIdx.x;
  const int rowBase = blk * ROWS;
  const int yBase   = blk * YROWS;
  const int lane = tid & 31;                  // wave32
  const int wave = tid >> 5;
  const int rg = wave & 3;                    // row group: 16 rows
  const int kh = wave >> 2;                   // K half: 512 cols

  // ---- one-time stage of this block's weight slices into LDS ----
#if HAVE_TDM
  if (tid < 32) {   // wave 0 issues the DMA (EXEC ignored by TDM)
    tdm_load_2d<0>((unsigned)(size_t)(void*)Ws, W + (size_t)rowBase * NRES,
                   NRES, ROWS, NRES, ROWS, NRES, /*padded=*/1);
    __builtin_amdgcn_s_wait_tensorcnt(0);
  }
#else
  for (int idx = tid; idx < ROWS * NRES; idx += NT) {
    int r = idx >> 10, k = idx & (NRES - 1);
    Ws[r * WSTRIDE + k + (k >> 8)] = W[(size_t)(rowBase + r) * NRES + k];
  }
#endif
  for (int idx = tid; idx < ROWS * NIN; idx += NT) {
    int r = idx >> 6, j = idx & 63;
    Wins[r * SSTRIDE + j] = W_in[(rowBase + r) * NIN + j];
    Wfbs[r * SSTRIDE + j] = W_fb[(rowBase + r) * NOUT + j];
  }
  for (int idx = tid; idx < YROWS * NCAT; idx += NT)
    Wouts[idx] = W_out[(size_t)yBase * NCAT + idx];
  for (int idx = tid; idx < NRES; idx += NT) st[idx] = laststate[idx];
  if (tid < NOUT) yprev[tid] = lastoutput[tid];
  __syncthreads();

  // A-matrix 16x4 f32 layout: lanes 0-15 -> K 0,1 ; lanes 16-31 -> K 2,3
  const int m = lane & 15;
  const int kOff = (lane >> 4) * 2;
  const float* wrow = &Ws[(rg * 16 + m) * WSTRIDE];

  for (int t = 0; t < T_STEPS; ++t) {
    // phase 1: bring in u_t
    if (tid < NIN) u_s[tid] = inputs[(size_t)t * NIN + tid];
    __syncthreads();

    // phase 2: bias = W_in @ u + W_feedb @ y_prev  (64 rows, one thread each)
    if (tid < ROWS) {
      float b = 0.f;
      #pragma unroll 8
      for (int j = 0; j < NIN; ++j)
        b += Wins[tid * SSTRIDE + j] * u_s[j] + Wfbs[tid * SSTRIDE + j] * yprev[j];
      bias[tid] = b;
    }

    // phase 3: pre-rows = Wslice @ state via fp32 WMMA (EXEC all ones)
    {
      const int kBase = kh * 512;
#if HAVE_WMMA_F32
      v8f acc = {};
      #pragma unroll 4
      for (int k = kBase; k < kBase + 512; k += 4) {
        const int k2 = k + (k >> 8) + kOff;   // TDM-padded row layout
        v2f a, bv;
        a.x  = wrow[k2];           // A[m, kOff..kOff+1]
        a.y  = wrow[k2 + 1];
        bv.x = st[k + kOff];       // B[k, n] = state[k]  (broadcast over N)
        bv.y = st[k + kOff + 1];
        acc = __builtin_amdgcn_wmma_f32_16x16x4_f32(
            false, a, false, bv, (short)0, acc, false, false);
      }
      // D layout: lane n<16 VGPR v = C[M=v, N=n]; lane>=16 -> M=v+8.
      // All columns are equal; lanes 0 and 16 publish the 16 row sums.
      if (lane == 0) {
        #pragma unroll
        for (int i = 0; i < 8; ++i) part[kh * ROWS + rg * 16 + i] = acc[i];
      } else if (lane == 16) {
        #pragma unroll
        for (int i = 0; i < 8; ++i) part[kh * ROWS + rg * 16 + 8 + i] = acc[i];
      }
#else
      float accs = 0.f;
      for (int k = kBase; k < kBase + 512; k += 4) {
        const int k2 = k + (k >> 8) + kOff;
        accs += wrow[k2]     * st[k + kOff] +
                wrow[k2 + 1] * st[k + kOff + 1];
      }
      accs += __shfl_xor(accs, 16, 32);   // combine K 0,1 with K 2,3
      if (lane < 16) part[kh * ROWS + rg * 16 + lane] = accs;
#endif
    }
    __syncthreads();

    // phase 4: s = tanh(pre) + noise ; publish this block's 64 state rows
    if (tid < ROWS) {
      float pre = part[tid] + part[ROWS + tid] + bias[tid];
      float s = tanhf(pre) + noise[(size_t)t * NRES + rowBase + tid];
      stateG[rowBase + tid] = s;
    }
    grid_barrier(cnt, gen);   // state fully visible device-wide

    // phase 6: refresh LDS state (TDM broadcast), compute 4 readout rows
#if HAVE_TDM
    if (tid < 32) {
      tdm_load_2d<16>((unsigned)(size_t)(void*)st, stateG,  // scope=DEV
                      NRES, 1, NRES, 1, NRES, /*padded=*/0);
      __builtin_amdgcn_s_wait_tensorcnt(0);
    }
    __syncthreads();
#else
    for (int i = tid; i < NRES; i += NT) st[i] = stateG[i];
    __syncthreads();
#endif
    {
      const int g = tid >> 6, j = tid & 63;
      float a2 = 0.f;
      for (int jj = j; jj < NCAT; jj += 64) {
        float v = (jj < NRES) ? st[jj] : u_s[jj - NRES];
        a2 += Wouts[g * NCAT + jj] * v;
      }
      red[tid] = a2;
    }
    __syncthreads();
    if (tid < YROWS) {
      float d = 0.f;
      #pragma unroll 8
      for (int i = 0; i < 64; ++i) d += red[tid * 64 + i];
      float yv = tanhf(d);                         // y_t (feedback value)
      yG[yBase + tid] = yv;
      out[(size_t)t * NOUT + yBase + tid] = tanhf(yv);  // predict(): tanh(ys)
    }
    grid_barrier(cnt, gen);   // y fully visible device-wide

    // phase 8: pull feedback vector for next step
    if (tid < NOUT) yprev[tid] = yG[tid];
  }
}

extern "C" void kernel_launch(void* const* d_in, const int* in_sizes, int n_in,
                              void* d_out, int out_size, void* d_ws, size_t ws_size,
                              hipStream_t stream) {
  (void)in_sizes; (void)n_in; (void)out_size; (void)ws_size;
  const float* inputs  = (const float*)d_in[0];
  const float* W       = (const float*)d_in[1];
  const float* W_in    = (const float*)d_in[2];
  const float* W_fb    = (const float*)d_in[3];
  const float* W_out   = (const float*)d_in[4];
  const float* noise   = (const float*)d_in[5];
  const float* lastst  = (const float*)d_in[6];
  const float* lastout = (const float*)d_in[7];
  float* out = (float*)d_out;

  float* stateG = (float*)d_ws;            // 1024 floats
  float* yG     = stateG + NRES;           // 64 floats
  unsigned* cnt = (unsigned*)(yG + NOUT);  // barrier count
  unsigned* gen = cnt + 1;                 // barrier generation

  (void)hipFuncSetAttribute((const void*)esn_kernel,
                            hipFuncAttributeMaxDynamicSharedMemorySize,
                            SMEM_FLOATS * (int)sizeof(float));

  esn_init_kernel<<<1, 1, 0, stream>>>(cnt, gen);
  esn_kernel<<<dim3(NB), dim3(NT), SMEM_FLOATS * sizeof(float), stream>>>(
      inputs, W, W_in, W_fb, W_out, noise, lastst, lastout,
      out, stateG, yG, cnt, gen);
}